// FMREDynamicDropout_44624710206140
// MI455X (gfx1250) — compile-verified
//
#include <hip/hip_runtime.h>
#include <stdint.h>

// FMREDynamicDropout for MI455X (gfx1250, wave32).
// Memory-bound streaming op: 308MB traffic -> ~13.2us floor @ 23.3 TB/s.
// Threefry-2x32 (JAX key(42) => keypair (0,42)) computed in-register,
// overlapped with NT b128 loads. Per-channel Bernoulli threshold is
// precomputed as an exact integer mantissa threshold (u < k  <=>
// (bits>>9) < ceil(k*2^23)), broadcast via CDNA5 async global->LDS DMA.

typedef float v4f __attribute__((ext_vector_type(4)));

#define TF_ROUND(r)                                     \
    {                                                   \
        x0 += x1;                                       \
        x1 = (x1 << (r)) | (x1 >> (32 - (r)));          \
        x1 ^= x0;                                       \
    }

// threefry2x32 with key (0, 42) == jax.random.key(42)
__device__ __forceinline__ void threefry2x32_0_42(uint32_t c0, uint32_t c1,
                                                  uint32_t& o0, uint32_t& o1) {
    const uint32_t ks0 = 0u;
    const uint32_t ks1 = 42u;
    const uint32_t ks2 = 0x1BD11BDAu ^ ks0 ^ ks1;
    uint32_t x0 = c0 + ks0;
    uint32_t x1 = c1 + ks1;
    TF_ROUND(13) TF_ROUND(15) TF_ROUND(26) TF_ROUND(6)
    x0 += ks1; x1 += ks2 + 1u;
    TF_ROUND(17) TF_ROUND(29) TF_ROUND(16) TF_ROUND(24)
    x0 += ks2; x1 += ks0 + 2u;
    TF_ROUND(13) TF_ROUND(15) TF_ROUND(26) TF_ROUND(6)
    x0 += ks0; x1 += ks1 + 3u;
    TF_ROUND(17) TF_ROUND(29) TF_ROUND(16) TF_ROUND(24)
    x0 += ks1; x1 += ks2 + 4u;
    TF_ROUND(13) TF_ROUND(15) TF_ROUND(26) TF_ROUND(6)
    x0 += ks2; x1 += ks0 + 5u;
    o0 = x0;
    o1 = x1;
}

// --- Kernel 1: min/max over feature_importance, emit integer thresholds ----
// JAX: u = bitcast((bits>>9)|0x3F800000) - 1 = (bits>>9) * 2^-23 (exact).
// mask = u < keep  <=>  (bits>>9) < ceil(keep * 2^23)   (exact, fp64 ceil).
__global__ __launch_bounds__(512) void FMRE_thresh_kernel(
    const float* __restrict__ fi, uint32_t* __restrict__ thr, int C) {
    __shared__ float smin[512];
    __shared__ float smax[512];
    const int t = threadIdx.x;
    const float v = fi[(t < C) ? t : 0];  // identity-safe filler
    smin[t] = v;
    smax[t] = v;
    __syncthreads();
#pragma unroll
    for (int s = 256; s > 0; s >>= 1) {
        if (t < s) {
            smin[t] = fminf(smin[t], smin[t + s]);
            smax[t] = fmaxf(smax[t], smax[t + s]);
        }
        __syncthreads();
    }
    if (t < C) {
        const float mn = smin[0];
        const float mx = smax[0];
        const float scaled = (v - mn) / (mx - mn);
        // keep = 1 - (0.1 + 0.4*(1-scaled)) = 0.5 + 0.4*scaled
        const float keep = 0.5f + 0.4f * scaled;
        thr[t] = (uint32_t)ceil((double)keep * 8388608.0);  // keep * 2^23
    }
}

// --- Kernel 2: mask + multiply over paired halves --------------------------
__global__ __launch_bounds__(256) void FMRE_dropout_kernel(
    const float* __restrict__ x, const uint32_t* __restrict__ thr_g,
    float* __restrict__ out, unsigned quads, unsigned half) {
    constexpr unsigned C = 384;    // channels (fixed harness shape)
    constexpr unsigned QPC = 784;  // (56*56)/4 quads per (b,c) image plane
    __shared__ uint32_t sthr[C];

    const unsigned t = threadIdx.x;
    // Stage thresholds into LDS via CDNA5 async global->LDS DMA (96 x b128).
    if (t < C / 4u) {
        const uint32_t lds_off =
            (uint32_t)(uintptr_t)(&sthr[0]) + t * 16u;  // LDS byte offset
        const uint32_t* gsrc = thr_g + (size_t)t * 4u;
        asm volatile("global_load_async_to_lds_b128 %0, %1, off"
                     :
                     : "v"(lds_off), "v"(gsrc)
                     : "memory");
    }
    asm volatile("s_wait_asynccnt 0" ::: "memory");
    __syncthreads();

    const unsigned q = blockIdx.x * 256u + t;  // quad id in first half
    if (q >= quads) return;

    const unsigned e0 = q * 4u;     // element index, first half
    const unsigned e1 = e0 + half;  // paired element, second half

    // Streaming (footprint > 192MB L2): non-temporal b128 loads.
    const v4f a = __builtin_nontemporal_load((const v4f*)(x + e0));
    const v4f b = __builtin_nontemporal_load((const v4f*)(x + e1));

    // Channel of each quad (HW=3136 is a multiple of 4 -> constant per quad).
    const unsigned c0 = (q / QPC) % C;
    const unsigned c1 = ((q + quads) / QPC) % C;
    const uint32_t T0 = sthr[c0];  // same-address LDS read broadcasts in-wave
    const uint32_t T1 = sthr[c1];

    // JAX pairing: out[i] = hash(i, i+n/2).x ; out[i+n/2] = .y
    uint32_t r0a, r0b, r1a, r1b, r2a, r2b, r3a, r3b;
    threefry2x32_0_42(e0 + 0u, e1 + 0u, r0a, r0b);
    threefry2x32_0_42(e0 + 1u, e1 + 1u, r1a, r1b);
    threefry2x32_0_42(e0 + 2u, e1 + 2u, r2a, r2b);
    threefry2x32_0_42(e0 + 3u, e1 + 3u, r3a, r3b);

    v4f oa, ob;
    oa.x = ((r0a >> 9) < T0) ? a.x : 0.0f;
    oa.y = ((r1a >> 9) < T0) ? a.y : 0.0f;
    oa.z = ((r2a >> 9) < T0) ? a.z : 0.0f;
    oa.w = ((r3a >> 9) < T0) ? a.w : 0.0f;
    ob.x = ((r0b >> 9) < T1) ? b.x : 0.0f;
    ob.y = ((r1b >> 9) < T1) ? b.y : 0.0f;
    ob.z = ((r2b >> 9) < T1) ? b.z : 0.0f;
    ob.w = ((r3b >> 9) < T1) ? b.w : 0.0f;

    __builtin_nontemporal_store(oa, (v4f*)(out + e0));
    __builtin_nontemporal_store(ob, (v4f*)(out + e1));
}

extern "C" void kernel_launch(void* const* d_in, const int* in_sizes, int n_in,
                              void* d_out, int out_size, void* d_ws,
                              size_t ws_size, hipStream_t stream) {
    (void)n_in;
    (void)out_size;
    (void)ws_size;
    const float* x = (const float*)d_in[0];
    const float* fi = (const float*)d_in[1];
    float* out = (float*)d_out;
    uint32_t* thr = (uint32_t*)d_ws;  // C uint32 thresholds in scratch

    const int C = in_sizes[1];                 // 384
    const unsigned n = (unsigned)in_sizes[0];  // 32*384*56*56 = 38,535,168
    const unsigned half = n / 2u;              // 19,267,584 (mult of 4)
    const unsigned quads = half / 4u;          // 4,816,896

    FMRE_thresh_kernel<<<1, 512, 0, stream>>>(fi, thr, C);

    const unsigned blocks = (quads + 255u) / 256u;  // 18,816
    FMRE_dropout_kernel<<<blocks, 256, 0, stream>>>(x, thr, out, quads, half);
}